// VectorizedODT_81552839016748
// MI455X (gfx1250) — compile-verified
//
#include <hip/hip_runtime.h>
#include <hip/hip_bf16.h>

// Problem constants (from reference)
#define NB   16      // datasets
#define NS   1024    // samples
#define NF   256     // features
#define NT   64      // trees
#define ND   6       // depth
#define NC   64      // leaves (2^6)
#define NU   128     // units
#define TD   (NT*ND) // 384 columns of the selector GEMM

typedef __attribute__((ext_vector_type(2))) float v2f;
typedef __attribute__((ext_vector_type(8))) float v8f;

// ---------------------------------------------------------------------------
// Kernel 1: column softmax over F=256 of feature_logits [F, T, D] -> selT[col][f]
// (stored transposed so WMMA B-tiles read contiguous K pairs)
// ---------------------------------------------------------------------------
__global__ void k_softmax_sel(const float* __restrict__ logits,
                              float* __restrict__ selT) {
    __shared__ float red[NF];
    const int col = blockIdx.x;        // t*6+d, 0..383
    const int f   = threadIdx.x;       // 0..255
    const float v = logits[(size_t)f * TD + col];
    red[f] = v;
    __syncthreads();
    for (int s = NF / 2; s > 0; s >>= 1) {
        if (f < s) red[f] = fmaxf(red[f], red[f + s]);
        __syncthreads();
    }
    const float m = red[0];
    __syncthreads();
    const float e = expf(v - m);
    red[f] = e;
    __syncthreads();
    for (int s = NF / 2; s > 0; s >>= 1) {
        if (f < s) red[f] += red[f + s];
        __syncthreads();
    }
    selT[(size_t)col * NF + f] = e / red[0];
}

// ---------------------------------------------------------------------------
// Kernel 2: FV = X @ sel via V_WMMA_F32_16X16X4_F32, fused sigmoid epilogue.
// One wave = one 16x16 tile of FV.  Writes P[d][s][col] = sigmoid((fv-thr)*exp(-lt)).
// grid: (NS/16, TD/64, nDatasets), block: 128 (4 waves -> 4 N-tiles)
// ---------------------------------------------------------------------------
__global__ void k_fv_sigmoid(const float* __restrict__ X,
                             const float* __restrict__ selT,
                             const float* __restrict__ thr,
                             const float* __restrict__ ltemp,
                             float* __restrict__ P) {
    const int d    = blockIdx.z;
    const int lane = threadIdx.x & 31;
    const int wv   = threadIdx.x >> 5;
    const int half = lane >> 4;        // 0: K pair {0,1}; 1: K pair {2,3}
    const int srow = lane & 15;        // A row / B column within tile
    const int m0   = blockIdx.x * 16;
    const int ncol = (blockIdx.y * 4 + wv) * 16 + srow;   // 0..383

    const float* xrow = X + ((size_t)d * NS + (m0 + srow)) * NF;
    const float* brow = selT + (size_t)ncol * NF;

    v8f acc = {};
    for (int k0 = 0; k0 < NF; k0 += 4) {
        const int ka = k0 + 2 * half;
        v2f a; a.x = xrow[ka]; a.y = xrow[ka + 1];
        v2f b; b.x = brow[ka]; b.y = brow[ka + 1];
        acc = __builtin_amdgcn_wmma_f32_16x16x4_f32(false, a, false, b,
                                                    (short)0, acc, false, false);
    }

    const float th = thr[ncol];
    const float sc = expf(-ltemp[ncol]);
    float* Pd = P + (size_t)d * NS * TD;
#pragma unroll
    for (int r = 0; r < 8; ++r) {
        const int M = r + 8 * half;
        const float tl = (acc[r] - th) * sc;
        Pd[(size_t)(m0 + M) * TD + ncol] = 1.0f / (1.0f + expf(-tl));
    }
}

// ---------------------------------------------------------------------------
// Kernel 3: leaf weights + tree-summed response GEMM, fully fused.
// Block = 256 threads (8 waves) -> out tile [16 samples, 128 units].
// Per tree: cooperative w[16][64] into LDS, then 16 WMMA f32 16x16x4 steps.
// grid: (NS/16, nDatasets)
// ---------------------------------------------------------------------------
#define SPST 385   // padded stride for p (385 % 64 == 1)
#define SWST 65    // padded stride for w (65 % 64 == 1)

__global__ void k_trees_out(const float* __restrict__ P,
                            const float* __restrict__ resp,
                            float* __restrict__ out) {
    __shared__ float sp[16 * SPST];   // sigmoid probs: 16 samples x 384 cols
    __shared__ float sw[16 * SWST];   // leaf weights:  16 samples x 64 leaves

    const int d    = blockIdx.y;
    const int tid  = threadIdx.x;
    const int lane = tid & 31;
    const int wv   = tid >> 5;         // 0..7 -> unit tile
    const int half = lane >> 4;
    const int srow = lane & 15;
    const int m0   = blockIdx.x * 16;
    const int ucol = wv * 16 + srow;   // unit 0..127

    const float* Pd = P + (size_t)d * NS * TD;
    for (int i = tid; i < 16 * TD; i += 256) {
        const int s = i / TD, c = i - s * TD;
        sp[s * SPST + c] = Pd[(size_t)(m0 + s) * TD + c];
    }
    __syncthreads();

    v8f acc = {};
    for (int t = 0; t < NT; ++t) {
        // cooperative leaf weights for this tree (shared A-tile for all 8 waves)
        for (int i = tid; i < 16 * NC; i += 256) {
            const int s = i >> 6, c = i & 63;
            const float* pp = &sp[s * SPST + t * ND];
            float prod = 1.0f;
#pragma unroll
            for (int dd = 0; dd < ND; ++dd) {
                const float pv = pp[dd];
                prod *= ((c >> dd) & 1) ? pv : (1.0f - pv);
            }
            sw[s * SWST + c] = prod;
        }
        __syncthreads();

        const float* rrow = resp + ((size_t)t * NU + ucol) * NC; // [t][u][c], c contiguous
        const float* arow = &sw[srow * SWST];
#pragma unroll
        for (int k0 = 0; k0 < NC; k0 += 4) {
            const int ka = k0 + 2 * half;
            v2f a; a.x = arow[ka]; a.y = arow[ka + 1];
            v2f b; b.x = rrow[ka]; b.y = rrow[ka + 1];
            acc = __builtin_amdgcn_wmma_f32_16x16x4_f32(false, a, false, b,
                                                        (short)0, acc, false, false);
        }
        __syncthreads();   // protect sw before next tree rewrites it
    }

    float* od = out + (size_t)d * NS * NU;
#pragma unroll
    for (int r = 0; r < 8; ++r) {
        const int M = r + 8 * half;
        od[(size_t)(m0 + M) * NU + ucol] = acc[r] * (1.0f / (float)NT);
    }
}

// ---------------------------------------------------------------------------
extern "C" void kernel_launch(void* const* d_in, const int* in_sizes, int n_in,
                              void* d_out, int out_size, void* d_ws, size_t ws_size,
                              hipStream_t stream) {
    const float* x    = (const float*)d_in[0];  // [16,1024,256]
    const float* fl   = (const float*)d_in[1];  // [256,64,6]
    const float* thr  = (const float*)d_in[2];  // [64,6]
    const float* lt   = (const float*)d_in[3];  // [64,6]
    const float* resp = (const float*)d_in[4];  // [64,128,64]
    float* out = (float*)d_out;                 // [16,1024,128]

    char* ws = (char*)d_ws;
    float* selT = (float*)ws;                               // 384*256 f32
    const size_t selT_bytes = (size_t)TD * NF * sizeof(float);
    const size_t poff = (selT_bytes + 255) & ~(size_t)255;
    float* P = (float*)(ws + poff);
    const size_t P_full = (size_t)NB * NS * TD * sizeof(float);   // ~25 MB

    k_softmax_sel<<<TD, NF, 0, stream>>>(fl, selT);

    if (ws_size >= poff + P_full) {
        // batched path: one launch covers all 16 datasets
        k_fv_sigmoid<<<dim3(NS / 16, TD / 64, NB), 128, 0, stream>>>(x, selT, thr, lt, P);
        k_trees_out<<<dim3(NS / 16, NB), 256, 0, stream>>>(P, resp, out);
    } else {
        // small-workspace fallback: reuse one per-dataset P buffer (stream-ordered)
        for (int ds = 0; ds < NB; ++ds) {
            k_fv_sigmoid<<<dim3(NS / 16, TD / 64, 1), 128, 0, stream>>>(
                x + (size_t)ds * NS * NF, selT, thr, lt, P);
            k_trees_out<<<dim3(NS / 16, 1), 256, 0, stream>>>(
                P, resp, out + (size_t)ds * NS * NU);
        }
    }
}